// ChebGCN2_50371376447888
// MI455X (gfx1250) — compile-verified
//
#include <hip/hip_runtime.h>

typedef float v2f __attribute__((ext_vector_type(2)));
typedef float v8f __attribute__((ext_vector_type(8)));

static constexpr int F_IN  = 96;
static constexpr int F_HID = 64;
static constexpr int F_OUT = 40;

// ---------------------------------------------------------------------------
// Zero-fill (re-run every launch: deg / tx1 / tx2 are accumulated into)
// ---------------------------------------------------------------------------
__global__ void zero_kernel(float* __restrict__ p, long n) {
    long i = (long)blockIdx.x * blockDim.x + threadIdx.x;
    long stride = (long)gridDim.x * blockDim.x;
    for (; i < n; i += stride) p[i] = 0.0f;
}

// ---------------------------------------------------------------------------
// Degree accumulation: deg[row[e]] += 1   (hardware f32 atomic)
// ---------------------------------------------------------------------------
__global__ void degree_kernel(const int* __restrict__ row, int E,
                              float* __restrict__ deg) {
    int i = blockIdx.x * blockDim.x + threadIdx.x;
    if (i < E) unsafeAtomicAdd(&deg[row[i]], 1.0f);
}

// ---------------------------------------------------------------------------
// deg_inv_sqrt = deg > 0 ? rsqrt(deg) : 0
// ---------------------------------------------------------------------------
__global__ void dinv_kernel(const float* __restrict__ deg,
                            float* __restrict__ dis, int n) {
    int i = blockIdx.x * blockDim.x + threadIdx.x;
    if (i < n) {
        float d = deg[i];
        dis[i] = (d > 0.0f) ? rsqrtf(d) : 0.0f;
    }
}

// ---------------------------------------------------------------------------
// Edge scatter:  dst[row[e], :] += (-dis[row]*dis[col]) * src[col, :]
// One wave32 per edge; lane l covers features l, l+32, ... (coalesced loads
// and coalesced global_atomic_add_f32).
// ---------------------------------------------------------------------------
template <int F>
__global__ void scatter_kernel(const float* __restrict__ src,
                               const int* __restrict__ row,
                               const int* __restrict__ col,
                               const float* __restrict__ dis,
                               float* __restrict__ dst, int E) {
    constexpr int FPL = F / 32;  // features per lane
    int lane   = threadIdx.x & 31;
    int wave   = (blockIdx.x * blockDim.x + threadIdx.x) >> 5;
    int nwaves = (gridDim.x * blockDim.x) >> 5;
    for (int e = wave; e < E; e += nwaves) {
        int r = row[e];
        int c = col[e];
        float norm = -dis[r] * dis[c];
        const float* s = src + (long)c * F;
        float*       d = dst + (long)r * F;
#pragma unroll
        for (int j = 0; j < FPL; ++j) {
            int f = lane + 32 * j;
            unsafeAtomicAdd(&d[f], norm * s[f]);
        }
    }
}

// ---------------------------------------------------------------------------
// Fused dual-GEMM + bias (+ReLU):  Dst = act(A1@B1 + A2@B2 + bias)
// fp32 WMMA 16x16x4, one wave per 16x16 output tile.
//
// Fragment layouts (CDNA5 ISA §7.12.2, wave32):
//   A 16x4 : lane = 16*half + m ; a[v] = A[m][k0 + 2*half + v]
//   B 4x16 : lane = 16*half + n ; b[v] = B[k0 + 2*half + v][n]
//   C 16x16: vgpr v, lane = 16*half + n ; C[v + 8*half][n]
// EXEC stays all-ones through every WMMA (uniform wave-level early exit only;
// N-padding handled branchlessly; store mask applied after the last WMMA).
// ---------------------------------------------------------------------------
template <int K, int NOUT, bool RELU>
__global__ __launch_bounds__(256)
void wmma_dual_gemm(const float* __restrict__ A1,   // [M, K]
                    const float* __restrict__ A2,   // [M, K]
                    const float* __restrict__ B1,   // [K, NOUT]
                    const float* __restrict__ B2,   // [K, NOUT]
                    const float* __restrict__ bias, // [NOUT]
                    float* __restrict__ Dst,        // [M, NOUT]
                    int M) {
    constexpr int NTILES = (NOUT + 15) / 16;
    int wave   = (blockIdx.x * blockDim.x + threadIdx.x) >> 5;
    int mtiles = M >> 4;
    int total  = mtiles * NTILES;
    if (wave >= total) return;  // uniform per wave -> EXEC all-ones below

    int mt = wave / NTILES;
    int nt = wave % NTILES;
    int m0 = mt << 4;
    int n0 = nt << 4;

    int lane = threadIdx.x & 31;
    int half = lane >> 4;       // 0 or 1
    int ml   = lane & 15;       // A-fragment row / B,C column index
    int n    = n0 + ml;
    bool ncol_ok = (n < NOUT);

    v8f c = {};

    // --- term 1: A1 @ B1 ---
    {
        const float* arow = A1 + (long)(m0 + ml) * K;
        for (int k0 = 0; k0 < K; k0 += 4) {
            int kb = k0 + 2 * half;
            v2f a, b;
            a.x = arow[kb];
            a.y = arow[kb + 1];
            b.x = ncol_ok ? B1[(long)kb * NOUT + n]       : 0.0f;
            b.y = ncol_ok ? B1[(long)(kb + 1) * NOUT + n] : 0.0f;
            c = __builtin_amdgcn_wmma_f32_16x16x4_f32(
                    false, a, false, b, (short)0, c, false, false);
        }
    }
    // --- term 2: A2 @ B2 (accumulates into same c) ---
    {
        const float* arow = A2 + (long)(m0 + ml) * K;
        for (int k0 = 0; k0 < K; k0 += 4) {
            int kb = k0 + 2 * half;
            v2f a, b;
            a.x = arow[kb];
            a.y = arow[kb + 1];
            b.x = ncol_ok ? B2[(long)kb * NOUT + n]       : 0.0f;
            b.y = ncol_ok ? B2[(long)(kb + 1) * NOUT + n] : 0.0f;
            c = __builtin_amdgcn_wmma_f32_16x16x4_f32(
                    false, a, false, b, (short)0, c, false, false);
        }
    }

    float bv = ncol_ok ? bias[n] : 0.0f;
#pragma unroll
    for (int v = 0; v < 8; ++v) {
        int rowm = m0 + v + 8 * half;
        float val = c[v] + bv;
        if (RELU) val = fmaxf(val, 0.0f);
        if (ncol_ok && rowm < M) Dst[(long)rowm * NOUT + n] = val;
    }
}

// ---------------------------------------------------------------------------
// Host-side orchestration
// ---------------------------------------------------------------------------
extern "C" void kernel_launch(void* const* d_in, const int* in_sizes, int n_in,
                              void* d_out, int out_size, void* d_ws, size_t ws_size,
                              hipStream_t stream) {
    const float* x    = (const float*)d_in[0];
    const int*   ei   = (const int*)d_in[1];
    const float* W1_0 = (const float*)d_in[2];
    const float* W1_1 = (const float*)d_in[3];
    const float* b1   = (const float*)d_in[4];
    const float* W2_0 = (const float*)d_in[5];
    const float* W2_1 = (const float*)d_in[6];
    const float* b2   = (const float*)d_in[7];
    float* out = (float*)d_out;

    const int N = in_sizes[0] / F_IN;   // 50000
    const int E = in_sizes[1] / 2;      // 800000
    const int* row = ei;
    const int* col = ei + E;

    // Workspace layout (accumulated buffers first so one zero pass covers them):
    //   [deg N] [tx1 N*F_IN] [tx2 N*F_HID] [dis N] [h N*F_HID]
    float* deg = (float*)d_ws;
    float* tx1 = deg + N;
    float* tx2 = tx1 + (size_t)N * F_IN;
    float* dis = tx2 + (size_t)N * F_HID;
    float* h   = dis + N;

    // 1) zero deg + tx1 + tx2
    long nzero = (long)N * (1 + F_IN + F_HID);
    zero_kernel<<<2048, 256, 0, stream>>>(deg, nzero);

    // 2) degree
    degree_kernel<<<(E + 255) / 256, 256, 0, stream>>>(row, E, deg);

    // 3) deg_inv_sqrt
    dinv_kernel<<<(N + 255) / 256, 256, 0, stream>>>(deg, dis, N);

    // 4) tx1 = -D^-1/2 A D^-1/2 x   (one wave per edge)
    {
        int waves_per_block = 256 / 32;
        int blocks = (E + waves_per_block - 1) / waves_per_block;
        scatter_kernel<F_IN><<<blocks, 256, 0, stream>>>(x, row, col, dis, tx1, E);
    }

    // 5) h = relu(x@W1_0 + tx1@W1_1 + b1)
    {
        int tiles = (N / 16) * ((F_HID + 15) / 16);  // 3125 * 4
        int blocks = (tiles * 32 + 255) / 256;
        wmma_dual_gemm<F_IN, F_HID, true><<<blocks, 256, 0, stream>>>(
            x, tx1, W1_0, W1_1, b1, h, N);
    }

    // 6) tx2 = -D^-1/2 A D^-1/2 h
    {
        int waves_per_block = 256 / 32;
        int blocks = (E + waves_per_block - 1) / waves_per_block;
        scatter_kernel<F_HID><<<blocks, 256, 0, stream>>>(h, row, col, dis, tx2, E);
    }

    // 7) out = h@W2_0 + tx2@W2_1 + b2
    {
        int tiles = (N / 16) * ((F_OUT + 15) / 16);  // 3125 * 3
        int blocks = (tiles * 32 + 255) / 256;
        wmma_dual_gemm<F_HID, F_OUT, false><<<blocks, 256, 0, stream>>>(
            h, tx2, W2_0, W2_1, b2, out, N);
    }
}